// GatTimeDecoder_38620345925667
// MI455X (gfx1250) — compile-verified
//
#include <hip/hip_runtime.h>
#include <math.h>

// Problem constants (from reference setup): B=8, K=256, F=64, E=128
#define BB 8
#define KK 256
#define FF 64
#define EE 128
#define NROWS (BB * KK)     // 2048
#define N1   (2 * EE)       // 256 columns of the fused projection
#define ALPHA 0.2f

typedef float v2f __attribute__((ext_vector_type(2)));
typedef float v8f __attribute__((ext_vector_type(8)));

// Only check on the gfx1250 device pass — the host (x86) pass reports 0 for
// all __builtin_amdgcn_* in __has_builtin.
#if defined(__gfx1250__) && !__has_builtin(__builtin_amdgcn_wmma_f32_16x16x4_f32)
#error "missing __builtin_amdgcn_wmma_f32_16x16x4_f32 on gfx1250 device pass"
#endif

__device__ __forceinline__ v8f wmma_f32_k4(v2f a, v2f b, v8f c) {
  // D = A(16x4,f32) x B(4x16,f32) + C(16x16,f32)
  return __builtin_amdgcn_wmma_f32_16x16x4_f32(
      /*neg_a=*/false, a, /*neg_b=*/false, b,
      /*c_mod=*/(short)0, c, /*reuse_a=*/false, /*reuse_b=*/false);
}

// ---------------------------------------------------------------------------
// Kernel 1: S(2048x256) = X(2048x64) @ Wcat^T(64x256), +lin_b folded into the
// s_i half (cols 0..127). Wcat row e'<128 -> w1[e'] = lin_w[e'][0:64];
// e'>=128 -> w2[e'-128] = lin_w[e'-128][64:128].
// One wave per 16x16 tile; K=64 in steps of 4 -> 16 v_wmma per tile.
// ---------------------------------------------------------------------------
__global__ void k1_proj_gemm(const float* __restrict__ x,
                             const float* __restrict__ lin_w,
                             const float* __restrict__ lin_b,
                             float* __restrict__ S) {
  const int wave  = (blockIdx.x * blockDim.x + threadIdx.x) >> 5;
  const int lane  = threadIdx.x & 31;
  const int tileN = wave & 15;   // 256/16 tiles across
  const int tileM = wave >> 4;   // 2048/16 tiles down
  const int half  = lane >> 4;
  const int ll    = lane & 15;

  const int m = tileM * 16 + ll;          // A row held by this lane
  const int n = tileN * 16 + ll;          // B column held by this lane
  const float* __restrict__ arow = x + m * FF;
  const float* __restrict__ wrow =
      (n < EE) ? (lin_w + n * (2 * FF)) : (lin_w + (n - EE) * (2 * FF) + FF);

  v8f c = {0.f, 0.f, 0.f, 0.f, 0.f, 0.f, 0.f, 0.f};
#pragma unroll
  for (int k0 = 0; k0 < FF; k0 += 4) {
    const int kk = k0 + 2 * half;
    v2f a = *(const v2f*)(arow + kk);     // A[m][kk], A[m][kk+1]
    v2f b = *(const v2f*)(wrow + kk);     // B[kk][n], B[kk+1][n] (= Wcat[n][kk..])
    c = wmma_f32_k4(a, b, c);
  }
  const float bias = (n < EE) ? lin_b[n] : 0.f;   // fold lin_b into s_i half
#pragma unroll
  for (int v = 0; v < 8; ++v) {
    const int row = tileM * 16 + v + 8 * half;    // C/D row layout
    S[row * N1 + n] = c[v] + bias;
  }
}

// ---------------------------------------------------------------------------
// Kernel 2 (VALU): per (b,i) row, e_j = sum_e a_e * leaky(u_e + v_je) + bias,
// then softmax over j. One wave per (b,i); 8 j's per lane; u and a cached in
// LDS; __shfl_xor tree reductions (wave32) for max/sum.
// ---------------------------------------------------------------------------
__global__ void k2_scores_softmax(const float* __restrict__ S,
                                  const float* __restrict__ avec,
                                  const float* __restrict__ att_bias,
                                  float* __restrict__ ATT) {
  __shared__ __align__(16) float uS[8][EE];
  __shared__ __align__(16) float aS[EE];

  const int w    = threadIdx.x >> 5;
  const int lane = threadIdx.x & 31;
  const int g = blockIdx.x * 8 + w;      // global row 0..2047
  const int b = g >> 8;
  const int i = g & 255;

  if (threadIdx.x < EE) aS[threadIdx.x] = avec[threadIdx.x];
  // u = s_i row (lin_b already folded in)
  *(float4*)(&uS[w][lane * 4]) = *(const float4*)(S + g * N1 + lane * 4);
  __syncthreads();

  const float* __restrict__ SJ = S + (b * KK) * N1 + EE;  // s_j half
  float ev[8];
#pragma unroll
  for (int t = 0; t < 8; ++t) {
    const int j = lane + 32 * t;
    const float* __restrict__ vj = SJ + j * N1;
    float acc = 0.f;
#pragma unroll 4
    for (int e = 0; e < EE; e += 4) {
      float4 vv = *(const float4*)(vj + e);
      float4 uu = *(const float4*)(&uS[w][e]);
      float4 aa = *(const float4*)(&aS[e]);
      float z0 = uu.x + vv.x; acc += aa.x * (z0 >= 0.f ? z0 : ALPHA * z0);
      float z1 = uu.y + vv.y; acc += aa.y * (z1 >= 0.f ? z1 : ALPHA * z1);
      float z2 = uu.z + vv.z; acc += aa.z * (z2 >= 0.f ? z2 : ALPHA * z2);
      float z3 = uu.w + vv.w; acc += aa.w * (z3 >= 0.f ? z3 : ALPHA * z3);
    }
    ev[t] = acc + att_bias[i * KK + j];
  }

  // softmax over 256 values (8 per lane, wave32 cross-lane reduce)
  float mx = ev[0];
#pragma unroll
  for (int t = 1; t < 8; ++t) mx = fmaxf(mx, ev[t]);
#pragma unroll
  for (int off = 16; off >= 1; off >>= 1) mx = fmaxf(mx, __shfl_xor(mx, off, 32));
  float s = 0.f;
#pragma unroll
  for (int t = 0; t < 8; ++t) { ev[t] = __expf(ev[t] - mx); s += ev[t]; }
#pragma unroll
  for (int off = 16; off >= 1; off >>= 1) s += __shfl_xor(s, off, 32);
  const float inv = 1.f / s;
#pragma unroll
  for (int t = 0; t < 8; ++t) ATT[g * KK + lane + 32 * t] = ev[t] * inv;
}

// ---------------------------------------------------------------------------
// Kernel 3 (fused): per 16-row strip (within one batch):
//   H(16x64)  = sigmoid(Att rows(16x256) @ X_b(256x64))     [WMMA, K=256]
//   staged in LDS (padded against bank conflicts), then
//   Out(16x64) = H(16x64) @ fc_w^T(64x64) + fc_b            [WMMA, K=64]
// Block = 128 threads = 4 waves, wave w owns 16 output columns.
// ---------------------------------------------------------------------------
__global__ void k3_apply_fc(const float* __restrict__ ATT,
                            const float* __restrict__ x,
                            const float* __restrict__ fc_w,
                            const float* __restrict__ fc_b,
                            float* __restrict__ out) {
  __shared__ __align__(16) float Hs[16][FF + 4];  // +4 pad: column reads hit distinct banks

  const int w    = threadIdx.x >> 5;
  const int lane = threadIdx.x & 31;
  const int half = lane >> 4;
  const int ll   = lane & 15;
  const int r0 = blockIdx.x * 16;        // strip of 16 global rows
  const int b  = r0 >> 8;
  const int ncol = w * 16 + ll;          // output column owned by this lane

  // ---- Stage 1: H tile ----
  const float* __restrict__ Arow = ATT + (r0 + ll) * KK;
  const float* __restrict__ Xb   = x + b * KK * FF;
  v8f c = {0.f, 0.f, 0.f, 0.f, 0.f, 0.f, 0.f, 0.f};
  for (int k0 = 0; k0 < KK; k0 += 4) {
    const int kk = k0 + 2 * half;
    v2f a = *(const v2f*)(Arow + kk);
    v2f bv;
    bv.x = Xb[kk * FF + ncol];
    bv.y = Xb[(kk + 1) * FF + ncol];
    c = wmma_f32_k4(a, bv, c);
  }
#pragma unroll
  for (int v = 0; v < 8; ++v) {
    const int row = v + 8 * half;
    Hs[row][ncol] = 1.f / (1.f + __expf(-c[v]));   // sigmoid
  }
  __syncthreads();

  // ---- Stage 2: Out tile = Hs @ fc_w^T + fc_b ----
  const float* __restrict__ frow = fc_w + ncol * FF;  // fc_w[ncol][*], contiguous K
  v8f d = {0.f, 0.f, 0.f, 0.f, 0.f, 0.f, 0.f, 0.f};
#pragma unroll
  for (int k0 = 0; k0 < FF; k0 += 4) {
    const int kk = k0 + 2 * half;
    v2f a;
    a.x = Hs[ll][kk];
    a.y = Hs[ll][kk + 1];
    v2f bv = *(const v2f*)(frow + kk);
    d = wmma_f32_k4(a, bv, d);
  }
  const float bias = fc_b[ncol];
#pragma unroll
  for (int v = 0; v < 8; ++v) {
    const int row = r0 + v + 8 * half;
    out[row * FF + ncol] = d[v] + bias;
  }
}

// ---------------------------------------------------------------------------
// Launch
// ---------------------------------------------------------------------------
extern "C" void kernel_launch(void* const* d_in, const int* in_sizes, int n_in,
                              void* d_out, int out_size, void* d_ws, size_t ws_size,
                              hipStream_t stream) {
  const float* x        = (const float*)d_in[0];  // (8,256,64)
  const float* lin_w    = (const float*)d_in[1];  // (128,128)
  const float* lin_b    = (const float*)d_in[2];  // (128,)
  const float* avec     = (const float*)d_in[3];  // (128,1)
  const float* att_bias = (const float*)d_in[4];  // (256,256)
  const float* fc_w     = (const float*)d_in[5];  // (64,64)
  const float* fc_b     = (const float*)d_in[6];  // (64,)
  float* out = (float*)d_out;                     // (8,256,64)

  float* S   = (float*)d_ws;                      // 2048 x 256 f32 (2 MB)
  float* ATT = S + (size_t)NROWS * N1;            // 2048 x 256 f32 (2 MB)

  // K1: 2048x256 output -> 2048 tiles of 16x16, 4 waves/block -> 512 blocks
  k1_proj_gemm<<<512, 128, 0, stream>>>(x, lin_w, lin_b, S);
  // K2: one wave per (b,i) row, 8 waves/block -> 256 blocks
  k2_scores_softmax<<<256, 256, 0, stream>>>(S, avec, att_bias, ATT);
  // K3: one 16-row strip per block -> 128 blocks, 4 waves each
  k3_apply_fc<<<128, 128, 0, stream>>>(ATT, x, fc_w, fc_b, out);
}